// Dense3DSpatialTransformer_82214263980284
// MI455X (gfx1250) — compile-verified
//
#include <hip/hip_runtime.h>

// Dense 3D spatial transformer (trilinear warp), B=2 C=2 H=W=D=160.
// Memory-bound gather: ~230 MB effective HBM traffic -> ~10us floor @ 23.3 TB/s.
// No matmul structure -> WMMA inapplicable; the CDNA5 paths that matter are the
// async global->LDS staging (ASYNCcnt), wide B64 gathers/stores, wave32 blocks.
// This revision processes a z-adjacent voxel PAIR per thread: halves index math
// per voxel, scalarizes y, makes stores B64.

constexpr int Hh = 160, Ww = 160, Dd = 160;
constexpr int WD  = Ww * Dd;            // 25600
constexpr int HWD = Hh * WD;            // 4,096,000
constexpr int Bb = 2, Cc = 2;
constexpr int THREADS = 256;
constexpr int VPB = 2 * THREADS;        // 512 voxels per block
constexpr int BLOCKS_PER_BATCH = HWD / VPB;  // 8000 (exact)
constexpr int BLOCKS_PER_ROW   = WD / VPB;   // 50   (exact) -> y uniform per block

// 8-byte pair with only 4-byte alignment guarantee (z index has arbitrary parity).
struct __attribute__((packed, aligned(4))) f2 { float a, b; };

#if defined(__HIP_DEVICE_COMPILE__)
typedef __attribute__((address_space(1))) int gint;   // global
typedef __attribute__((address_space(3))) int lint;   // LDS
#endif

// Trilinear sample of both channels at (fy, fx, fz); paired-z B64 gathers.
__device__ __forceinline__ void sample2(const float* __restrict__ img0,
                                        const float* __restrict__ img1,
                                        float fy, float fx, float fz,
                                        float& o0, float& o1) {
  const int x0 = (int)floorf(fx);
  const int y0 = (int)floorf(fy);
  const int z0 = (int)floorf(fz);

  const int x0c = min(max(x0,     0), Ww - 1);
  const int x1c = min(max(x0 + 1, 0), Ww - 1);
  const int y0c = min(max(y0,     0), Hh - 1);
  const int y1c = min(max(y0 + 1, 0), Hh - 1);
  const int z0c = min(max(z0,     0), Dd - 1);
  const int z1c = min(max(z0 + 1, 0), Dd - 1);

  const float wx = (float)x1c - fx;
  const float wy = (float)y1c - fy;
  const float wz = (float)z1c - fz;

  const float w00 = wx * wy;
  const float w10 = wx * (1.f - wy);
  const float w01 = (1.f - wx) * wy;
  const float w11 = (1.f - wx) * (1.f - wy);
  const float wz1 = 1.f - wz;

  // z-pair gather: p in [0, D-2] always in-bounds; selects handle clamp collapse.
  const int  p      = min(z0c, Dd - 2);
  const bool selHi0 = (z0c != p);
  const bool selHi1 = (z1c != p);

  const int i00 = y0c * WD + x0c * Dd + p;
  const int i10 = y1c * WD + x0c * Dd + p;
  const int i01 = y0c * WD + x1c * Dd + p;
  const int i11 = y1c * WD + x1c * Dd + p;

  auto zlerp = [&](const float* base, int idx) -> float {
    f2 v = *(const f2*)(base + idx);
    float t0 = selHi0 ? v.b : v.a;
    float t1 = selHi1 ? v.b : v.a;
    return wz * t0 + wz1 * t1;
  };

  o0 = w00 * zlerp(img0, i00) + w10 * zlerp(img0, i10)
     + w01 * zlerp(img0, i01) + w11 * zlerp(img0, i11);
  o1 = w00 * zlerp(img1, i00) + w10 * zlerp(img1, i10)
     + w01 * zlerp(img1, i01) + w11 * zlerp(img1, i11);
}

__global__ __launch_bounds__(THREADS)
void warp3d_kernel(const float* __restrict__ img,
                   const float* __restrict__ flow,
                   float* __restrict__ out) {
  __shared__ float sflow[3 * VPB];   // [dy | dx | dz], 512 floats each

  const int t     = threadIdx.x;
  const int bid   = blockIdx.x;
  const int batch = bid / BLOCKS_PER_BATCH;
  const int rbid  = bid - batch * BLOCKS_PER_BATCH;
  const int y       = rbid / BLOCKS_PER_ROW;                 // scalar per block
  const int remBase = (rbid - y * BLOCKS_PER_ROW) * VPB;     // scalar per block
  const int s0      = rbid * VPB;

  const float* flowb = flow + (size_t)batch * 3 * HWD + s0;

  // ---- Stage the 3 flow channels (3 x 2KB contiguous) into LDS ----
#if defined(__HIP_DEVICE_COMPILE__) && defined(__gfx1250__) && \
    __has_builtin(__builtin_amdgcn_global_load_async_to_lds_b32) && \
    __has_builtin(__builtin_amdgcn_s_wait_asynccnt)
  {
#pragma unroll
    for (int c = 0; c < 3; ++c) {
      gint* gA = (gint*)(flowb + c * HWD + t);
      gint* gB = (gint*)(flowb + c * HWD + THREADS + t);
      lint* lA = (lint*)(sflow + c * VPB + t);
      lint* lB = (lint*)(sflow + c * VPB + THREADS + t);
      __builtin_amdgcn_global_load_async_to_lds_b32(gA, lA, 0, 0);
      __builtin_amdgcn_global_load_async_to_lds_b32(gB, lB, 0, 0);
    }
    __builtin_amdgcn_s_wait_asynccnt(0);
  }
#else
#pragma unroll
  for (int c = 0; c < 3; ++c) {
    sflow[c * VPB + t]           = flowb[c * HWD + t];
    sflow[c * VPB + THREADS + t] = flowb[c * HWD + THREADS + t];
  }
#endif
  __syncthreads();

  // ---- Voxel pair coordinates: s = s0 + 2t (even), s+1 shares y and x ----
  const int rem = remBase + 2 * t;        // even
  const int xq  = rem / Dd;               // one per-thread divide
  const int z   = rem - xq * Dd;          // even -> z+1 stays in the same row

  const float2 dy = *(const float2*)&sflow[2 * t];            // flow[:,0] pair
  const float2 dx = *(const float2*)&sflow[VPB + 2 * t];      // flow[:,1] pair
  const float2 dz = *(const float2*)&sflow[2 * VPB + 2 * t];  // flow[:,2] pair

  const float* img0 = img + (size_t)batch * Cc * HWD;  // channel 0 plane
  const float* img1 = img0 + HWD;                      // channel 1 plane

  float a0, a1, b0, b1;
  sample2(img0, img1, dy.x + (float)y, dx.x + (float)xq, dz.x + (float)z,       a0, a1);
  sample2(img0, img1, dy.y + (float)y, dx.y + (float)xq, dz.y + (float)(z + 1), b0, b1);

  float* outb = out + (size_t)batch * Cc * HWD;
  const int s = s0 + 2 * t;                       // even -> 8B-aligned stores
  *(float2*)&outb[s]       = make_float2(a0, b0); // channel 0
  *(float2*)&outb[HWD + s] = make_float2(a1, b1); // channel 1
}

extern "C" void kernel_launch(void* const* d_in, const int* in_sizes, int n_in,
                              void* d_out, int out_size, void* d_ws, size_t ws_size,
                              hipStream_t stream) {
  const float* image = (const float*)d_in[0];  // (B, C, H, W, D) f32
  const float* flow  = (const float*)d_in[1];  // (B, 3, H, W, D) f32
  float* out = (float*)d_out;                  // (B, C, H, W, D) f32

  dim3 grid(Bb * BLOCKS_PER_BATCH);            // 16000 blocks
  dim3 block(THREADS);
  warp3d_kernel<<<grid, block, 0, stream>>>(image, flow, out);
}